// ComGCN_84851373900029
// MI455X (gfx1250) — compile-verified
//
#include <hip/hip_runtime.h>
#include <hip/hip_bf16.h>
#include <math.h>

// ---------------------------------------------------------------------------
// ComGCN pipeline for MI455X (gfx1250, wave32, WMMA)
//   W       = max(lstm_fwd(conv_weight), lstm_bwd(conv_weight))      [64x64]
//   deg     = 1 + segment_sum(edge_weight, col);  dinv = rsqrt(deg)
//   Xw      = X @ W                      (WMMA f32 16x16x4)
//   X_ma    = scatter_add(Xw[row] * dinv[row]*w*dinv[col], col) + Xw*dinv^2
//   out     = (node_emb + com_emb + X_ma) @ fusion_W^T + fusion_b (WMMA)
// ---------------------------------------------------------------------------

typedef __attribute__((ext_vector_type(2))) float v2f;
typedef __attribute__((ext_vector_type(8))) float v8f;

#define C64 64

__device__ __forceinline__ float sigm(float x) { return 1.0f / (1.0f + __expf(-x)); }

__device__ __forceinline__ void atomic_add_f32(float* p, float v) {
    __hip_atomic_fetch_add(p, v, __ATOMIC_RELAXED, __HIP_MEMORY_SCOPE_AGENT);
}

// --- (a) tiny LSTM over the 64 rows of conv_weight --------------------------
__global__ void lstm_weight_kernel(const float* __restrict__ cw,
                                   const float* __restrict__ wih_f,
                                   const float* __restrict__ bih_f,
                                   const float* __restrict__ bhh_f,
                                   const float* __restrict__ wih_b,
                                   const float* __restrict__ bih_b,
                                   const float* __restrict__ bhh_b,
                                   float* __restrict__ Wout) {
    __shared__ float zf[4 * C64];
    __shared__ float zb[4 * C64];
    __shared__ float xr[C64];
    const int r = blockIdx.x;    // row of conv_weight, 0..63
    const int j = threadIdx.x;   // gate column, 0..255
    if (j < C64) xr[j] = cw[r * C64 + j];
    __syncthreads();
    float sf = 0.0f, sb = 0.0f;
    #pragma unroll 8
    for (int k = 0; k < C64; ++k) {
        const float x = xr[k];
        sf = fmaf(x, wih_f[j * C64 + k], sf);
        sb = fmaf(x, wih_b[j * C64 + k], sb);
    }
    zf[j] = sf + bih_f[j] + bhh_f[j];
    zb[j] = sb + bih_b[j] + bhh_b[j];
    __syncthreads();
    if (j < C64) {
        // torch gate order i, f, g, o ; c0 = 0 so f-gate is dead
        const float cf = sigm(zf[j]) * tanhf(zf[2 * C64 + j]);
        const float hf = sigm(zf[3 * C64 + j]) * tanhf(cf);
        const float cb = sigm(zb[j]) * tanhf(zb[2 * C64 + j]);
        const float hb = sigm(zb[j]) * tanhf(cb);
        Wout[r * C64 + j] = fmaxf(hf, hb);
    }
}

// --- (b) init: deg = 1 (self-loop weight), X_ma = 0 -------------------------
__global__ void init_kernel(float* __restrict__ deg, float* __restrict__ Xma,
                            int N, long long totNC) {
    const long long i = (long long)blockIdx.x * blockDim.x + threadIdx.x;
    if (i < totNC) Xma[i] = 0.0f;
    if (i < N) deg[i] = 1.0f;
}

// --- (c) degree accumulation ------------------------------------------------
__global__ void deg_kernel(const int* __restrict__ ei, const float* __restrict__ ew,
                           float* __restrict__ deg, int E) {
    const int e = blockIdx.x * blockDim.x + threadIdx.x;
    if (e < E) {
        const int col = ei[E + e];  // edge_index[1][e]
        atomic_add_f32(&deg[col], ew[e]);
    }
}

// --- (d) deg -> dinv (in place) ---------------------------------------------
__global__ void dinv_kernel(float* __restrict__ deg, int N) {
    const int i = blockIdx.x * blockDim.x + threadIdx.x;
    if (i < N) {
        const float d = deg[i];
        deg[i] = (d > 0.0f) ? rsqrtf(d) : 0.0f;
    }
}

// --- (e)/(g) WMMA GEMM: [N x 64] @ [64 x 64] --------------------------------
// MODE 0: out = A0 @ Bsrc                         (Bsrc row-major [k][n])
// MODE 1: out = (A0+A1+A2+A3*dinv^2) @ Bsrc^T + bias   (Bsrc is fusion_W[n][k])
// One wave per 16-row stripe; four 16x16 f32 accumulator tiles per wave via
// v_wmma_f32_16x16x4_f32.  B is staged in LDS in a K-pair-interleaved layout:
//   Bs[((k>>1)*64 + n)*2 + (k&1)]  ==  B[k][n]
// so each lane's (B[kk][n], B[kk+1][n]) fragment is ONE contiguous 8-byte
// slot -> single ds_load_b64 straight into the even-aligned WMMA operand pair
// (no v_mov marshalling, half the DS instructions vs ds_load_2addr).
template <int MODE>
__global__ void gemm64_wmma(const float* __restrict__ A0,
                            const float* __restrict__ A1,
                            const float* __restrict__ A2,
                            const float* __restrict__ A3,
                            const float* __restrict__ dinv,
                            const float* __restrict__ Bsrc,
                            const float* __restrict__ bias,
                            float* __restrict__ out, int N) {
    __shared__ float Bs[C64 * C64];  // 16 KB of 320 KB WGP LDS
    const int tid = threadIdx.x;
    for (int i = tid; i < C64 * C64; i += blockDim.x) {
        const int k = i >> 6, n = i & 63;
        const float v = (MODE == 0) ? Bsrc[i] : Bsrc[n * C64 + k];  // B[k][n]
        Bs[(((k >> 1) * C64 + n) << 1) + (k & 1)] = v;
    }
    __syncthreads();

    const int wave = tid >> 5;
    const int lane = tid & 31;
    const int row0 = (blockIdx.x * (blockDim.x >> 5) + wave) * 16;
    if (row0 >= N) return;

    const int mrow  = row0 + (lane & 15);
    const int mc    = (mrow < N) ? mrow : (N - 1);      // clamp tail loads
    const int khalf = (lane >> 4) * 2;                  // ISA A 16x4 layout
    const int nlo   = lane & 15;

    float d2 = 0.0f;
    if (MODE == 1) {
        const float di = dinv[mc];
        d2 = di * di;                                   // self-loop norm
    }

    v8f acc[4] = {};
    #pragma unroll
    for (int k = 0; k < C64; k += 4) {
        const int kk = k + khalf;                       // lane's K pair base
        v2f a;
        if (MODE == 0) {
            a = *(const v2f*)&A0[(size_t)mc * C64 + kk];
        } else {
            const size_t o = (size_t)mc * C64 + kk;
            a.x = A0[o]     + A1[o]     + A2[o]     + A3[o]     * d2;
            a.y = A0[o + 1] + A1[o + 1] + A2[o + 1] + A3[o + 1] * d2;
        }
        #pragma unroll
        for (int t = 0; t < 4; ++t) {
            const v2f b = *(const v2f*)&Bs[((kk >> 1) * C64 + t * 16 + nlo) << 1];
            acc[t] = __builtin_amdgcn_wmma_f32_16x16x4_f32(
                false, a, false, b, (short)0, acc[t], false, false);
        }
    }

    // C/D layout: VGPR v -> row (row0 + v) for lanes 0-15, (row0 + 8 + v) for 16-31
    const int rbase = row0 + (lane >> 4) * 8;
    #pragma unroll
    for (int t = 0; t < 4; ++t) {
        const int col = t * 16 + nlo;
        const float bb = (MODE == 1) ? bias[col] : 0.0f;
        #pragma unroll
        for (int v = 0; v < 8; ++v) {
            const int r = rbase + v;
            if (r < N) out[(size_t)r * C64 + col] = acc[t][v] + bb;
        }
    }
}

// --- (f) edge gather/scale/scatter-add: one thread per (edge, 2 channels) ---
__global__ void scatter_kernel(const int* __restrict__ ei, const float* __restrict__ ew,
                               const float* __restrict__ dinv,
                               const float* __restrict__ Xw, float* __restrict__ Xma,
                               int E) {
    const long long gid = (long long)blockIdx.x * blockDim.x + threadIdx.x;
    const long long tot = (long long)E * (C64 / 2);
    if (gid >= tot) return;
    const int e = (int)(gid >> 5);
    const int c = (int)(gid & 31) * 2;
    const int row = ei[e];
    const int col = ei[E + e];
    const float norm = dinv[row] * ew[e] * dinv[col];
    const v2f xv = *(const v2f*)&Xw[(size_t)row * C64 + c];
    float* dst = &Xma[(size_t)col * C64 + c];
    atomic_add_f32(dst,     xv.x * norm);
    atomic_add_f32(dst + 1, xv.y * norm);
}

// ---------------------------------------------------------------------------
extern "C" void kernel_launch(void* const* d_in, const int* in_sizes, int n_in,
                              void* d_out, int out_size, void* d_ws, size_t ws_size,
                              hipStream_t stream) {
    const float* X        = (const float*)d_in[0];
    const int*   ei       = (const int*)d_in[1];   // edge_index [2,E]
    const float* ew       = (const float*)d_in[2];
    const float* node_emb = (const float*)d_in[3];
    const float* com_emb  = (const float*)d_in[4];
    const float* cw       = (const float*)d_in[5];
    const float* wih_f    = (const float*)d_in[6];
    const float* bih_f    = (const float*)d_in[7];
    const float* bhh_f    = (const float*)d_in[8];
    const float* wih_b    = (const float*)d_in[9];
    const float* bih_b    = (const float*)d_in[10];
    const float* bhh_b    = (const float*)d_in[11];
    const float* fW       = (const float*)d_in[12];
    const float* fb       = (const float*)d_in[13];
    float*       out      = (float*)d_out;

    const int N = in_sizes[0] / C64;
    const int E = in_sizes[2];

    // workspace layout (floats): W[4096] | deg/dinv[N] | Xw[N*64] | Xma[N*64]
    float* ws  = (float*)d_ws;
    float* W   = ws;
    float* deg = W + C64 * C64;
    float* Xw  = deg + N;
    float* Xma = Xw + (size_t)N * C64;

    const long long totNC = (long long)N * C64;
    const long long totEh = (long long)E * (C64 / 2);

    lstm_weight_kernel<<<C64, 256, 0, stream>>>(cw, wih_f, bih_f, bhh_f,
                                                wih_b, bih_b, bhh_b, W);
    init_kernel<<<(int)((totNC + 255) / 256), 256, 0, stream>>>(deg, Xma, N, totNC);
    deg_kernel<<<(E + 255) / 256, 256, 0, stream>>>(ei, ew, deg, E);
    dinv_kernel<<<(N + 255) / 256, 256, 0, stream>>>(deg, N);

    const int rowsPerBlock = 16 * (256 / 32);  // 8 waves * 16 rows
    const int gBlocks = (N + rowsPerBlock - 1) / rowsPerBlock;
    gemm64_wmma<0><<<gBlocks, 256, 0, stream>>>(X, nullptr, nullptr, nullptr,
                                                nullptr, W, nullptr, Xw, N);
    scatter_kernel<<<(int)((totEh + 255) / 256), 256, 0, stream>>>(ei, ew, deg, Xw, Xma, E);
    gemm64_wmma<1><<<gBlocks, 256, 0, stream>>>(node_emb, com_emb, Xma, Xw,
                                                deg, fW, fb, out, N);
}